// SubgraphAttention_2783138807992
// MI455X (gfx1250) — compile-verified
//
#include <hip/hip_runtime.h>
#include <math.h>

// ---------------------------------------------------------------------------
// SubgraphAttention for MI455X (gfx1250, wave32).
// B=32, N=65, S=64, D=512.
//   q = nve @ Wq^T + bq ; k = nve @ Wk^T + bk      (fp32 WMMA 16x16x4)
//   attn = softmax( (q k^T / sqrt(D)) * diag-mask )
//   greedy leader/member scan; padded expansion out[B,S,S+1,D].
// Output write (272 MB) dominates the roofline (~12us @ 23.3 TB/s), so the
// GEMMs stay in full fp32 via V_WMMA_F32_16X16X4_F32 (no precision gamble on
// the 0.05 threshold).
// ---------------------------------------------------------------------------

typedef float v2f __attribute__((ext_vector_type(2)));
typedef float v8f __attribute__((ext_vector_type(8)));

#define BB 32
#define NN 65
#define SS 64
#define DD 512

// --------------------------------------------------------------------------
// Kernel 1: q/k projection GEMM.  M = B*N = 2080 rows (130 tiles of 16),
// N = 512 cols (32 tiles of 16), K = 512 (128 wmma steps of 4).
// blockIdx.z: 0 -> q (Wq,bq), 1 -> k (Wk,bk).  One wave per tile.
// f32 A 16x4 layout: lane<16 -> M=lane, K pair {0,1}; lane>=16 -> K pair {2,3}.
// f32 B 4x16 layout mirrors it with N in place of M.
// --------------------------------------------------------------------------
__global__ __launch_bounds__(32)
void qk_gemm_kernel(const float* __restrict__ nve,
                    const float* __restrict__ Wq, const float* __restrict__ bq,
                    const float* __restrict__ Wk, const float* __restrict__ bk,
                    float* __restrict__ qout, float* __restrict__ kout)
{
    const int lane  = threadIdx.x;
    const int n0    = blockIdx.x * 16;
    const int m0    = blockIdx.y * 16;
    const bool isK  = (blockIdx.z != 0);

    const float* W    = isK ? Wk : Wq;
    const float* bias = isK ? bk : bq;
    float*       out  = isK ? kout : qout;

    const int l15   = lane & 15;
    const int khalf = (lane >> 4) * 2;          // 0 or 2

    const float* arow = nve + (size_t)(m0 + l15) * DD;   // A row (M contiguous in K)
    const float* brow = W   + (size_t)(n0 + l15) * DD;   // B[k][n] = W[n][k]

    v8f acc = {};
    for (int k = 0; k < DD; k += 4) {
        v2f a, b;
        a.x = arow[k + khalf];
        a.y = arow[k + khalf + 1];
        b.x = brow[k + khalf];
        b.y = brow[k + khalf + 1];
        acc = __builtin_amdgcn_wmma_f32_16x16x4_f32(
            /*neg_a=*/false, a, /*neg_b=*/false, b,
            /*c_mod=*/(short)0, acc, /*reuse_a=*/false, /*reuse_b=*/false);
    }

    // C/D layout: VGPR r, lanes 0-15 -> (M=r, N=lane); lanes 16-31 -> (M=8+r).
    const float bv    = bias[n0 + l15];
    const int   mbase = m0 + (lane >> 4) * 8;
#pragma unroll
    for (int r = 0; r < 8; ++r)
        out[(size_t)(mbase + r) * DD + (n0 + l15)] = acc[r] + bv;
}

// --------------------------------------------------------------------------
// Kernel 2: attention logits per batch: q[b] (65x512) @ k[b]^T (512x65),
// scaled by 1/sqrt(512); masked diagonal (i==j, i>=1) stored as exactly 0
// (reference multiplies the logit by 0 pre-softmax).  5x5 tiles of 16
// (padded to 80; OOB rows/cols read inside ws and are never stored).
// --------------------------------------------------------------------------
__global__ __launch_bounds__(32)
void attn_logits_kernel(const float* __restrict__ q,
                        const float* __restrict__ k,
                        float* __restrict__ attn)
{
    const int lane = threadIdx.x;
    const int n0   = blockIdx.x * 16;
    const int m0   = blockIdx.y * 16;
    const int b    = blockIdx.z;

    const int l15   = lane & 15;
    const int khalf = (lane >> 4) * 2;

    const float* qb = q + (size_t)b * NN * DD;
    const float* kb = k + (size_t)b * NN * DD;
    const float* arow = qb + (size_t)(m0 + l15) * DD;
    const float* brow = kb + (size_t)(n0 + l15) * DD;   // B[kk][n] = k[b][n][kk]

    v8f acc = {};
    for (int kk = 0; kk < DD; kk += 4) {
        v2f a, bm;
        a.x  = arow[kk + khalf];
        a.y  = arow[kk + khalf + 1];
        bm.x = brow[kk + khalf];
        bm.y = brow[kk + khalf + 1];
        acc = __builtin_amdgcn_wmma_f32_16x16x4_f32(
            false, a, false, bm, (short)0, acc, false, false);
    }

    const float scale = 0.044194173824159216f;  // 1/sqrt(512)
    const int   mbase = m0 + (lane >> 4) * 8;
    const int   j     = n0 + l15;
#pragma unroll
    for (int r = 0; r < 8; ++r) {
        const int i = mbase + r;
        if (i < NN && j < NN) {
            float v = acc[r] * scale;
            if (i == j && i >= 1) v = 0.0f;     // multiplicative mask pre-softmax
            attn[((size_t)b * NN + i) * NN + j] = v;
        }
    }
}

// --------------------------------------------------------------------------
// Kernel 3: row softmax over 65 elements; one wave32 per row.
// --------------------------------------------------------------------------
__global__ __launch_bounds__(32)
void softmax_kernel(float* __restrict__ attn)
{
    const int row  = blockIdx.x;         // 0 .. B*N-1
    const int lane = threadIdx.x;        // 0 .. 31
    float* a = attn + (size_t)row * NN;

    float x0 = a[lane];
    float x1 = a[lane + 32];
    float x2 = (lane == 0) ? a[64] : -1e30f;

    float m = fmaxf(fmaxf(x0, x1), x2);
    for (int off = 16; off > 0; off >>= 1)
        m = fmaxf(m, __shfl_xor(m, off, 32));

    float e0 = expf(x0 - m);
    float e1 = expf(x1 - m);
    float e2 = (lane == 0) ? expf(x2 - m) : 0.0f;

    float s = e0 + e1 + e2;
    for (int off = 16; off > 0; off >>= 1)
        s += __shfl_xor(s, off, 32);

    const float inv = 1.0f / s;
    a[lane]      = e0 * inv;
    a[lane + 32] = e1 * inv;
    if (lane == 0) a[64] = e2 * inv;
}

// --------------------------------------------------------------------------
// Kernel 4: greedy sequential clustering (the lax.scan).  One block of 64
// threads per batch; LDS `used` bitmap; 64 ordered steps.
//   members[i][j] = (attn[b,i+1,j+1] > 0.05) || (i==j)
//   active = !used[i];  sel = members & active;  used |= sel
// leader/member are written straight into their d_out slots as 0.0/1.0.
// --------------------------------------------------------------------------
__global__ __launch_bounds__(64)
void cluster_kernel(const float* __restrict__ attn,
                    float* __restrict__ leaderOut,    // [B,S]
                    float* __restrict__ memberOut)    // [B,S,S]
{
    const int b = blockIdx.x;
    const int j = threadIdx.x;           // 0..63
    __shared__ int used[SS];
    used[j] = 0;
    __syncthreads();

    for (int i = 0; i < SS; ++i) {
        const float aij = attn[((size_t)b * NN + (i + 1)) * NN + (j + 1)];
        const int memb = (aij > 0.05f) || (j == i);
        const int act  = !used[i];
        __syncthreads();                 // all reads of used[i] before updates
        const int sel = memb && act;
        memberOut[((size_t)b * SS + i) * SS + j] = sel ? 1.0f : 0.0f;
        if (j == i) leaderOut[b * SS + i] = act ? 1.0f : 0.0f;
        if (sel) used[j] = 1;            // only thread j writes used[j]
        __syncthreads();
    }
}

// --------------------------------------------------------------------------
// Kernel 5: padded expansion (the 272 MB streaming part).
//   out[b,i,0,:]   = leader[b,i] * cls
//   out[b,i,1+j,:] = member[b,i,j] * nve[b,1+j,:]
// One block per (b,i) row-group; 128 threads x float4 covers D=512.
// --------------------------------------------------------------------------
__global__ __launch_bounds__(128)
void expand_kernel(const float* __restrict__ nve,
                   const float* __restrict__ cls,
                   const float* __restrict__ leaderOut,
                   const float* __restrict__ memberOut,
                   float* __restrict__ out)
{
    const int bi = blockIdx.x;           // 0 .. B*S-1
    const int b  = bi >> 6;
    const int t  = threadIdx.x;          // 0..127 -> float4 lane over D

    const float  lead = leaderOut[bi];
    const float4 c4   = ((const float4*)cls)[t];

    float4* orow = (float4*)(out + (size_t)bi * (SS + 1) * DD);
    orow[t] = make_float4(lead * c4.x, lead * c4.y, lead * c4.z, lead * c4.w);

    const float* mrow = memberOut + (size_t)bi * SS;
    for (int s = 1; s <= SS; ++s) {
        const float  w = mrow[s - 1];
        const float4 v = ((const float4*)(nve + ((size_t)b * NN + s) * DD))[t];
        orow[(size_t)s * (DD / 4) + t] =
            make_float4(w * v.x, w * v.y, w * v.z, w * v.w);
    }
}

// --------------------------------------------------------------------------
extern "C" void kernel_launch(void* const* d_in, const int* in_sizes, int n_in,
                              void* d_out, int out_size, void* d_ws, size_t ws_size,
                              hipStream_t stream)
{
    (void)in_sizes; (void)n_in; (void)out_size; (void)ws_size;
    // setup_inputs order: desc_embeddings(unused), nve, Wq, bq, Wk, bk, cls
    const float* nve = (const float*)d_in[1];
    const float* Wq  = (const float*)d_in[2];
    const float* bq  = (const float*)d_in[3];
    const float* Wk  = (const float*)d_in[4];
    const float* bk  = (const float*)d_in[5];
    const float* cls = (const float*)d_in[6];

    float* out = (float*)d_out;

    // workspace: q | k | attn  (q,k adjacent so padded-tile reads stay in ws)
    float* qbuf = (float*)d_ws;
    float* kbuf = qbuf + (size_t)BB * NN * DD;               // 1,064,960 floats
    float* attn = kbuf + (size_t)BB * NN * DD;               // +135,200 floats

    // d_out = out [B,S,S+1,D] ++ leader [B,S] ++ member [B,S,S]
    float* leaderOut = out + (size_t)BB * SS * (SS + 1) * DD;
    float* memberOut = leaderOut + (size_t)BB * SS;

    dim3 g1(DD / 16, (BB * NN) / 16, 2);                     // 32 x 130 x {q,k}
    qk_gemm_kernel<<<g1, 32, 0, stream>>>(nve, Wq, bq, Wk, bk, qbuf, kbuf);

    dim3 g2((NN + 15) / 16, (NN + 15) / 16, BB);             // 5 x 5 x 32
    attn_logits_kernel<<<g2, 32, 0, stream>>>(qbuf, kbuf, attn);

    softmax_kernel<<<BB * NN, 32, 0, stream>>>(attn);

    cluster_kernel<<<BB, SS, 0, stream>>>(attn, leaderOut, memberOut);

    expand_kernel<<<BB * SS, 128, 0, stream>>>(nve, cls, leaderOut, memberOut, out);
}